// Pytorch3D_56538949484980
// MI455X (gfx1250) — compile-verified
//
#include <hip/hip_runtime.h>
#include <hip/hip_bf16.h>

// ---------------- problem constants (match reference) ----------------
constexpr int BB   = 4;
constexpr int VV   = 6890;
constexpr int FF   = 13776;
constexpr int FCH  = 256;              // faces per LDS chunk
constexpr int FPAD = 13824;            // FF rounded up to FCH multiple (54 chunks)
constexpr int NCH  = FPAD / FCH;       // 54
constexpr int Wp   = 224;
constexpr int Hp   = 224;
constexpr int HW   = Wp * Hp;          // 50176 = 512 * 98
constexpr int BLKS_PER_B = HW / 512;   // 98 (256 threads x 2 pixels)
constexpr float FEPS = 1e-8f;

// ---------------- CDNA5 async-to-LDS path (guarded) ----------------
typedef int v4i_ __attribute__((vector_size(16)));

#if __has_builtin(__builtin_amdgcn_global_load_async_to_lds_b128)
#define HAS_ASYNC 1
// builtin signature (from hipcc diagnostic): (v4i AS1*, v4i AS3*, imm int, imm int)
#define ASYNC_B128(gp, lp)                                                    \
  __builtin_amdgcn_global_load_async_to_lds_b128(                             \
      (__attribute__((address_space(1))) v4i_*)(gp),                          \
      (__attribute__((address_space(3))) v4i_*)(lp), 0, 0)
#else
#define HAS_ASYNC 0
#endif

#if __has_builtin(__builtin_amdgcn_s_wait_asynccnt)
#define WAIT_ASYNC(n) __builtin_amdgcn_s_wait_asynccnt(n)
#else
#define WAIT_ASYNC(n) asm volatile("s_wait_asynccnt %0" ::"i"(n) : "memory")
#endif

// ---------------- kernel 1: transform + project vertices ----------------
__global__ __launch_bounds__(256) void xform_kernel(
    const float* __restrict__ verts, const float* __restrict__ R,
    const float* __restrict__ T, float* __restrict__ xp,
    float* __restrict__ yp, float* __restrict__ zs) {
  int gid = blockIdx.x * 256 + threadIdx.x;
  if (gid >= BB * VV) return;
  int b = gid / VV;
  const float* Rb = R + b * 9;
  const float* Tb = T + b * 3;
  const float* vp = verts + (size_t)gid * 3;
  float vx = vp[0], vy = vp[1], vz = vp[2];
  // Tcam_i = sum_j R[i][j] * T[j]
  float tc0 = Rb[0] * Tb[0] + Rb[1] * Tb[1] + Rb[2] * Tb[2];
  float tc1 = Rb[3] * Tb[0] + Rb[4] * Tb[1] + Rb[5] * Tb[2];
  float tc2 = Rb[6] * Tb[0] + Rb[7] * Tb[1] + Rb[8] * Tb[2];
  // vcam_j = sum_i v_i * R[i][j] + Tcam_j
  float cx = vx * Rb[0] + vy * Rb[3] + vz * Rb[6] + tc0;
  float cy = vx * Rb[1] + vy * Rb[4] + vz * Rb[7] + tc1;
  float cz = vx * Rb[2] + vy * Rb[5] + vz * Rb[8] + tc2;
  float zsafe = (fabsf(cz) > FEPS) ? cz : FEPS;
  xp[gid] = 500.0f * cx / zsafe + (float)(Wp / 2);
  yp[gid] = 500.0f * cy / zsafe + (float)(Hp / 2);
  zs[gid] = zsafe;
}

// ---------------- kernel 2: per-face coefficient setup ----------------
// record (12 floats / 48B): e0x,e0y,e1x,e1y, x2,y2,iz0,iz1, iz2,0,0,0
__global__ __launch_bounds__(256) void facesetup_kernel(
    const int* __restrict__ faces, const float* __restrict__ xp,
    const float* __restrict__ yp, const float* __restrict__ zs,
    float* __restrict__ fd) {
  int gid = blockIdx.x * 256 + threadIdx.x;  // BB * FPAD threads exactly
  int b = gid / FPAD;
  int f = gid % FPAD;
  float4 r0 = make_float4(0.f, 0.f, 0.f, 0.f);
  float4 r1 = make_float4(0.f, 0.f, 0.f, 0.f);
  float4 r2 = make_float4(0.f, 0.f, 0.f, 0.f);
  if (f < FF) {
    int i0 = faces[f * 3 + 0], i1 = faces[f * 3 + 1], i2 = faces[f * 3 + 2];
    int base = b * VV;
    float x0 = xp[base + i0], x1 = xp[base + i1], x2 = xp[base + i2];
    float y0 = yp[base + i0], y1 = yp[base + i1], y2 = yp[base + i2];
    float z0 = zs[base + i0], z1 = zs[base + i1], z2 = zs[base + i2];
    float denom = (y1 - y2) * (x0 - x2) + (x2 - x1) * (y0 - y2);
    bool dok = fabsf(denom) > FEPS;
    bool ok = dok && (z0 > FEPS) && (z1 > FEPS) && (z2 > FEPS);
    float invd = 1.0f / (dok ? denom : 1.0f);
    r0 = make_float4((y1 - y2) * invd, (x2 - x1) * invd,
                     (y2 - y0) * invd, (x0 - x2) * invd);
    r1 = make_float4(x2, y2, ok ? 1.0f / z0 : 0.0f, ok ? 1.0f / z1 : 0.0f);
    r2 = make_float4(ok ? 1.0f / z2 : 0.0f, 0.f, 0.f, 0.f);
  }
  float4* o = (float4*)fd + (size_t)gid * 3;
  o[0] = r0; o[1] = r1; o[2] = r2;
}

// ---------------- kernel 3: rasterize (async LDS staged) ----------------
__global__ __launch_bounds__(256) void raster_kernel(
    const float* __restrict__ fd, float* __restrict__ zb,
    int* __restrict__ bidx) {
  __shared__ float4 sbuf[2][FCH * 3];  // 2 x 12KB double buffer
  const int t = threadIdx.x;
  const int b = blockIdx.x / BLKS_PER_B;
  const int blk = blockIdx.x % BLKS_PER_B;
  const int p0 = blk * 512 + t;
  const int p1 = p0 + 256;
  const float px0 = (float)(p0 % Wp) + 0.5f;
  const float py0 = (float)(p0 / Wp) + 0.5f;
  const float px1 = (float)(p1 % Wp) + 0.5f;
  const float py1 = (float)(p1 / Wp) + 0.5f;
  float binv0 = 0.0f, binv1 = 0.0f;   // best 1/z (larger = closer)
  int bi0 = -1, bi1 = -1;             // winning face index
  const float4* gbase = (const float4*)fd + (size_t)b * (FPAD * 3);

  auto process = [&](const float4* __restrict__ recs, int fbase) {
#pragma unroll 4
    for (int i = 0; i < FCH; ++i) {
      float4 A  = recs[i * 3 + 0];   // e0x,e0y,e1x,e1y
      float4 Bv = recs[i * 3 + 1];   // x2,y2,iz0,iz1
      float iz2 = recs[i * 3 + 2].x;
      {
        float ex = px0 - Bv.x, ey = py0 - Bv.y;
        float w0 = fmaf(A.x, ex, A.y * ey);
        float w1 = fmaf(A.z, ex, A.w * ey);
        float w2 = 1.0f - w0 - w1;
        float inv = fmaf(w0, Bv.z, fmaf(w1, Bv.w, w2 * iz2));
        bool ok = (w0 >= 0.0f) & (w1 >= 0.0f) & (w2 >= 0.0f) & (inv > binv0);
        binv0 = ok ? inv : binv0;
        bi0 = ok ? (fbase + i) : bi0;
      }
      {
        float ex = px1 - Bv.x, ey = py1 - Bv.y;
        float w0 = fmaf(A.x, ex, A.y * ey);
        float w1 = fmaf(A.z, ex, A.w * ey);
        float w2 = 1.0f - w0 - w1;
        float inv = fmaf(w0, Bv.z, fmaf(w1, Bv.w, w2 * iz2));
        bool ok = (w0 >= 0.0f) & (w1 >= 0.0f) & (w2 >= 0.0f) & (inv > binv1);
        binv1 = ok ? inv : binv1;
        bi1 = ok ? (fbase + i) : bi1;
      }
    }
  };

#if HAS_ASYNC
  {  // prologue: chunk 0 -> buffer 0 (3 x b128 per thread = one 48B record)
    const float4* g = gbase + t * 3;
    float4* l = &sbuf[0][t * 3];
    ASYNC_B128(g + 0, l + 0);
    ASYNC_B128(g + 1, l + 1);
    ASYNC_B128(g + 2, l + 2);
  }
  for (int c = 0; c < NCH; ++c) {
    const int cur = c & 1;
    if (c + 1 < NCH) {
      const float4* g = gbase + (size_t)(c + 1) * (FCH * 3) + t * 3;
      float4* l = &sbuf[cur ^ 1][t * 3];
      ASYNC_B128(g + 0, l + 0);
      ASYNC_B128(g + 1, l + 1);
      ASYNC_B128(g + 2, l + 2);
      WAIT_ASYNC(3);  // in-order: chunk c's 3 copies from this wave are done
    } else {
      WAIT_ASYNC(0);
    }
    __syncthreads();                 // all waves' copies of chunk c landed
    process(&sbuf[cur][0], c * FCH);
    __syncthreads();                 // reads done before buffer is overwritten
  }
#else
  for (int c = 0; c < NCH; ++c) {
    __syncthreads();
    const float4* g = gbase + (size_t)c * (FCH * 3) + t * 3;
    float4 a = g[0], d = g[1], e = g[2];
    sbuf[0][t * 3 + 0] = a;
    sbuf[0][t * 3 + 1] = d;
    sbuf[0][t * 3 + 2] = e;
    __syncthreads();
    process(&sbuf[0][0], c * FCH);
  }
#endif

  float z0 = (bi0 >= 0) ? (1.0f / binv0) : -1.0f;
  float z1 = (bi1 >= 0) ? (1.0f / binv1) : -1.0f;
  size_t o = (size_t)b * HW;
  zb[o + p0] = z0;
  zb[o + p1] = z1;
  bidx[o + p0] = bi0;
  bidx[o + p1] = bi1;
}

// ---------------- kernel 4: per-batch min/max of zbuf ----------------
__global__ __launch_bounds__(256) void minmax_kernel(const float* __restrict__ zb,
                                                     float* __restrict__ mm) {
  __shared__ float slo[256], shi[256];
  int b = blockIdx.x;
  int t = threadIdx.x;
  float lo = 3.4e38f, hi = -3.4e38f;
  for (int p = t; p < HW; p += 256) {
    float v = zb[(size_t)b * HW + p];
    lo = fminf(lo, v);
    hi = fmaxf(hi, v);
  }
  slo[t] = lo;
  shi[t] = hi;
  __syncthreads();
  for (int s = 128; s > 0; s >>= 1) {
    if (t < s) {
      slo[t] = fminf(slo[t], slo[t + s]);
      shi[t] = fmaxf(shi[t], shi[t + s]);
    }
    __syncthreads();
  }
  if (t == 0) {
    mm[b * 2 + 0] = slo[0];
    mm[b * 2 + 1] = shi[0];
  }
}

// ---------------- kernel 5: finalize to (B,5,H,W) ----------------
__global__ __launch_bounds__(256) void finalize_kernel(
    const float* __restrict__ zb, const int* __restrict__ bidx,
    const float* __restrict__ fcol, const float* __restrict__ mm,
    float* __restrict__ out) {
  int gid = blockIdx.x * 256 + threadIdx.x;  // BB*HW exactly
  int b = gid / HW;
  int p = gid % HW;
  float z = zb[gid];
  int idx = bidx[gid];
  float r = 0.f, g = 0.f, bl = 0.f;
  if (idx >= 0) {
    r = fcol[idx * 3 + 0];
    g = fcol[idx * 3 + 1];
    bl = fcol[idx * 3 + 2];
  }
  float mask = (z > -1.0f) ? 1.0f : 0.0f;
  float zmin = mm[b * 2 + 0], zmax = mm[b * 2 + 1];
  float depth = (z - zmin) / (zmax - zmin);
  size_t o = (size_t)b * 5 * HW + p;
  out[o + 0 * (size_t)HW] = r;
  out[o + 1 * (size_t)HW] = g;
  out[o + 2 * (size_t)HW] = bl;
  out[o + 3 * (size_t)HW] = mask;
  out[o + 4 * (size_t)HW] = depth;
}

// ---------------- host launcher ----------------
extern "C" void kernel_launch(void* const* d_in, const int* in_sizes, int n_in,
                              void* d_out, int out_size, void* d_ws,
                              size_t ws_size, hipStream_t stream) {
  (void)in_sizes; (void)n_in; (void)out_size; (void)ws_size;
  const float* verts = (const float*)d_in[0];  // (B,V,3)
  const float* fcol  = (const float*)d_in[1];  // (F,1,1,3)
  const float* R     = (const float*)d_in[2];  // (B,3,3)
  const float* T     = (const float*)d_in[3];  // (B,3)
  const int*   faces = (const int*)d_in[4];    // (F,3)
  float* out = (float*)d_out;

  float* ws = (float*)d_ws;
  float* xp = ws;                              // B*V
  float* yp = xp + (size_t)BB * VV;            // B*V
  float* zs = yp + (size_t)BB * VV;            // B*V
  float* fd = zs + (size_t)BB * VV;            // B*FPAD*12  (16B aligned)
  float* zb = fd + (size_t)BB * FPAD * 12;     // B*HW
  float* mm = zb + (size_t)BB * HW;            // 2*B
  int* bidx = (int*)(mm + 2 * BB);             // B*HW

  xform_kernel<<<(BB * VV + 255) / 256, 256, 0, stream>>>(verts, R, T, xp, yp, zs);
  facesetup_kernel<<<(BB * FPAD) / 256, 256, 0, stream>>>(faces, xp, yp, zs, fd);
  raster_kernel<<<BB * BLKS_PER_B, 256, 0, stream>>>(fd, zb, bidx);
  minmax_kernel<<<BB, 256, 0, stream>>>(zb, mm);
  finalize_kernel<<<(BB * HW) / 256, 256, 0, stream>>>(zb, bidx, fcol, mm, out);
}